// EventBuilder_71846212927739
// MI455X (gfx1250) — compile-verified
//
#include <hip/hip_runtime.h>
#include <math.h>

#define N_HITS   16384
#define N_EDGES  262144
#define N_GROUPS 512
#define IN_CH    25
#define HID      128
#define HEADS    4
#define DH       32
#define LAYERS   3
#define EDGE_DIM 5
#define FFN_DIM  512
#define LN_EPS   1e-5f
#define FKEY_NEG_INF 0x007FFFFFu

typedef __attribute__((ext_vector_type(2))) float v2f;
typedef __attribute__((ext_vector_type(8))) float v8f;

// ---------- helpers ----------
__device__ __forceinline__ unsigned fkey(float f) {
  unsigned u = __float_as_uint(f);
  return (u & 0x80000000u) ? ~u : (u | 0x80000000u);
}
__device__ __forceinline__ float fdecode(unsigned k) {
  unsigned u = (k & 0x80000000u) ? (k & 0x7FFFFFFFu) : ~k;
  return __uint_as_float(u);
}

// ---------- f32 WMMA GEMM, compile-time K/N, 32x64 tile per wave ----------
// C[M,NN] = A[M,KK] @ B[KK,NN] (+bias)(+relu).  One wave computes a 32x64 tile:
// MT=2 M-subtiles x NT=4 N-subtiles of 16x16 via V_WMMA_F32_16X16X4_F32.
// A-frag (ISA 7.12.2, 32-bit A 16x4): lane<16 -> M=lane, K=k0,k0+1 ;
// lane>=16 -> M=lane-16, K=k0+2,k0+3.  B rows striped across lanes with
// half-wave K split.  C/D: VGPR v -> M=v (lanes<16) / v+8 (lanes>=16).
// All loads use compile-time immediate offsets off fixed base pointers.
template<int KK, int NN, bool BIAS, bool RELU>
__global__ __launch_bounds__(32)
void gemm_wmma(const float* __restrict__ A, const float* __restrict__ B,
               const float* __restrict__ bias, float* __restrict__ C) {
  constexpr int MT = 2, NT = 4;
  const int m0   = blockIdx.x * (16 * MT);
  const int n0   = blockIdx.y * (16 * NT);
  const int lane = threadIdx.x;
  const int half = lane >> 4;
  const int lm   = lane & 15;

  v8f acc[MT][NT];
#pragma unroll
  for (int mt = 0; mt < MT; ++mt)
#pragma unroll
    for (int t = 0; t < NT; ++t) acc[mt][t] = (v8f){0.f,0.f,0.f,0.f,0.f,0.f,0.f,0.f};

  const float* arow0 = A + (size_t)(m0 + lm) * KK + half * 2;
  const float* arow1 = arow0 + (size_t)16 * KK;
  const float* bbase = B + (size_t)(half * 2) * NN + n0 + lm;

  constexpr int KFULL = (KK / 4) * 4;
#pragma unroll 4
  for (int k0 = 0; k0 < KFULL; k0 += 4) {
    v2f a0, a1;
    a0.x = arow0[k0];     a0.y = arow0[k0 + 1];
    a1.x = arow1[k0];     a1.y = arow1[k0 + 1];
    v2f bf[NT];
#pragma unroll
    for (int t = 0; t < NT; ++t) {
      bf[t].x = bbase[(size_t)k0 * NN + t * 16];
      bf[t].y = bbase[(size_t)(k0 + 1) * NN + t * 16];
    }
#pragma unroll
    for (int t = 0; t < NT; ++t) {
      acc[0][t] = __builtin_amdgcn_wmma_f32_16x16x4_f32(false, a0, false, bf[t],
                                                        (short)0, acc[0][t], false, false);
      acc[1][t] = __builtin_amdgcn_wmma_f32_16x16x4_f32(false, a1, false, bf[t],
                                                        (short)0, acc[1][t], false, false);
    }
  }
  if constexpr (KFULL < KK) {   // K tail (only embed K=25)
    constexpr int k0 = KFULL;
    const int ka = k0 + half * 2;
    v2f a0, a1;
    a0.x = (ka     < KK) ? arow0[k0]     : 0.f;
    a0.y = (ka + 1 < KK) ? arow0[k0 + 1] : 0.f;
    a1.x = (ka     < KK) ? arow1[k0]     : 0.f;
    a1.y = (ka + 1 < KK) ? arow1[k0 + 1] : 0.f;
#pragma unroll
    for (int t = 0; t < NT; ++t) {
      v2f b;
      b.x = (ka     < KK) ? bbase[(size_t)k0 * NN + t * 16]       : 0.f;
      b.y = (ka + 1 < KK) ? bbase[(size_t)(k0 + 1) * NN + t * 16] : 0.f;
      acc[0][t] = __builtin_amdgcn_wmma_f32_16x16x4_f32(false, a0, false, b,
                                                        (short)0, acc[0][t], false, false);
      acc[1][t] = __builtin_amdgcn_wmma_f32_16x16x4_f32(false, a1, false, b,
                                                        (short)0, acc[1][t], false, false);
    }
  }

#pragma unroll
  for (int mt = 0; mt < MT; ++mt) {
    const int mrow = m0 + mt * 16 + half * 8;
#pragma unroll
    for (int t = 0; t < NT; ++t) {
      float bs = 0.f;
      if (BIAS) bs = bias[n0 + t * 16 + lm];
      float* cc = C + (size_t)mrow * NN + n0 + t * 16 + lm;
#pragma unroll
      for (int v = 0; v < 8; ++v) {
        float val = acc[mt][t][v] + bs;
        if (RELU) val = fmaxf(val, 0.f);
        cc[(size_t)v * NN] = val;
      }
    }
  }
}

// ---------- fills ----------
__global__ void fill_u32(unsigned* __restrict__ p, unsigned v, int n) {
  int i = blockIdx.x * blockDim.x + threadIdx.x;
  if (i < n) p[i] = v;
}

// ---------- residual + layernorm (in place on h) ----------
__global__ __launch_bounds__(HID)
void add_ln_kernel(float* __restrict__ h, const float* __restrict__ delta,
                   const float* __restrict__ g, const float* __restrict__ b) {
  __shared__ float red[HID];
  const int row = blockIdx.x;
  const int c = threadIdx.x;
  float v = h[(size_t)row * HID + c] + delta[(size_t)row * HID + c];
  red[c] = v;
  __syncthreads();
  for (int s = HID / 2; s > 0; s >>= 1) {
    if (c < s) red[c] += red[c + s];
    __syncthreads();
  }
  float mu = red[0] * (1.f / HID);
  __syncthreads();
  float d = v - mu;
  red[c] = d * d;
  __syncthreads();
  for (int s = HID / 2; s > 0; s >>= 1) {
    if (c < s) red[c] += red[c + s];
    __syncthreads();
  }
  float var = red[0] * (1.f / HID);
  h[(size_t)row * HID + c] = d * rsqrtf(var + LN_EPS) * g[c] + b[c];
}

// ---------- edge attention: logits ----------
__global__ __launch_bounds__(256)
void edge_logits_kernel(const int* __restrict__ ei, const float* __restrict__ ea,
                        const float* __restrict__ qkv, const float* __restrict__ We,
                        const float* __restrict__ be, float* __restrict__ logits) {
  __shared__ float sWe[EDGE_DIM * HID];
  __shared__ float sbe[HID];
  for (int i = threadIdx.x; i < EDGE_DIM * HID; i += blockDim.x) sWe[i] = We[i];
  for (int i = threadIdx.x; i < HID; i += blockDim.x) sbe[i] = be[i];
  __syncthreads();
  const int idx = blockIdx.x * blockDim.x + threadIdx.x;
  if (idx >= N_EDGES * HEADS) return;
  const int e  = idx / HEADS;
  const int hd = idx % HEADS;
  const int src = ei[e];
  const int dst = ei[N_EDGES + e];
  const float a0 = ea[e * EDGE_DIM + 0], a1 = ea[e * EDGE_DIM + 1],
              a2 = ea[e * EDGE_DIM + 2], a3 = ea[e * EDGE_DIM + 3],
              a4 = ea[e * EDGE_DIM + 4];
  const int cb = hd * DH;
  const float* q = qkv + (size_t)dst * (3 * HID) + cb;
  const float* k = qkv + (size_t)src * (3 * HID) + HID + cb;
  float acc = 0.f;
#pragma unroll 4
  for (int d = 0; d < DH; ++d) {
    float ee = sbe[cb + d] + a0 * sWe[0 * HID + cb + d] + a1 * sWe[1 * HID + cb + d]
             + a2 * sWe[2 * HID + cb + d] + a3 * sWe[3 * HID + cb + d]
             + a4 * sWe[4 * HID + cb + d];
    acc += q[d] * (k[d] + ee);
  }
  logits[idx] = acc * 0.17677669529663687f;  // 1/sqrt(32)
}

// ---------- segment max over dst (uint-key atomicMax) ----------
__global__ __launch_bounds__(256)
void edge_segmax_kernel(const int* __restrict__ ei, const float* __restrict__ logits,
                        unsigned* __restrict__ mkey) {
  const int idx = blockIdx.x * blockDim.x + threadIdx.x;
  if (idx >= N_EDGES * HEADS) return;
  const int e  = idx / HEADS;
  const int hd = idx % HEADS;
  const int dst = ei[N_EDGES + e];
  atomicMax(&mkey[dst * HEADS + hd], fkey(logits[idx]));
}

// ---------- softmax numerator scatter: s += p, attn += p * (v[src]+e) ----------
__global__ __launch_bounds__(256)
void edge_accum_kernel(const int* __restrict__ ei, const float* __restrict__ ea,
                       const float* __restrict__ qkv, const float* __restrict__ We,
                       const float* __restrict__ be, const float* __restrict__ logits,
                       const unsigned* __restrict__ mkey, float* __restrict__ ssum,
                       float* __restrict__ attn) {
  __shared__ float sWe[EDGE_DIM * HID];
  __shared__ float sbe[HID];
  for (int i = threadIdx.x; i < EDGE_DIM * HID; i += blockDim.x) sWe[i] = We[i];
  for (int i = threadIdx.x; i < HID; i += blockDim.x) sbe[i] = be[i];
  __syncthreads();
  const int idx = blockIdx.x * blockDim.x + threadIdx.x;
  if (idx >= N_EDGES * HEADS) return;
  const int e  = idx / HEADS;
  const int hd = idx % HEADS;
  const int src = ei[e];
  const int dst = ei[N_EDGES + e];
  float m = fdecode(mkey[dst * HEADS + hd]);
  if (!isfinite(m)) m = 0.f;
  const float p = expf(logits[idx] - m);
  atomicAdd(&ssum[dst * HEADS + hd], p);
  const float a0 = ea[e * EDGE_DIM + 0], a1 = ea[e * EDGE_DIM + 1],
              a2 = ea[e * EDGE_DIM + 2], a3 = ea[e * EDGE_DIM + 3],
              a4 = ea[e * EDGE_DIM + 4];
  const int cb = hd * DH;
  const float* v = qkv + (size_t)src * (3 * HID) + 2 * HID + cb;
  float* ad = attn + (size_t)dst * HID + cb;
#pragma unroll 4
  for (int d = 0; d < DH; ++d) {
    float ee = sbe[cb + d] + a0 * sWe[0 * HID + cb + d] + a1 * sWe[1 * HID + cb + d]
             + a2 * sWe[2 * HID + cb + d] + a3 * sWe[3 * HID + cb + d]
             + a4 * sWe[4 * HID + cb + d];
    atomicAdd(&ad[d], p * (v[d] + ee));
  }
}

// ---------- attn /= max(s, 1e-9) ----------
__global__ __launch_bounds__(256)
void attn_div_kernel(float* __restrict__ attn, const float* __restrict__ ssum) {
  const int idx = blockIdx.x * blockDim.x + threadIdx.x;
  if (idx >= N_HITS * HID) return;
  const int n = idx / HID;
  const int c = idx % HID;
  attn[idx] /= fmaxf(ssum[n * HEADS + c / DH], 1e-9f);
}

// ---------- group pooling scatter ----------
__global__ __launch_bounds__(256)
void pool_scatter_kernel(const float* __restrict__ h, const float* __restrict__ x,
                         const int* __restrict__ gidx, unsigned* __restrict__ gmaxkey,
                         float* __restrict__ sumx, float* __restrict__ sumy,
                         float* __restrict__ cnt_all, float* __restrict__ cntx,
                         float* __restrict__ cnty) {
  const int idx = blockIdx.x * blockDim.x + threadIdx.x;
  if (idx >= N_HITS * HID) return;
  const int n = idx / HID;
  const int c = idx % HID;
  const int g = gidx[n];
  const float val = h[idx];
  atomicMax(&gmaxkey[g * HID + c], fkey(val));
  const int view = (int)x[(size_t)n * IN_CH + 3];  // trunc toward zero, matches astype(int32)
  if (view == 0) atomicAdd(&sumx[g * HID + c], val);
  else if (view == 1) atomicAdd(&sumy[g * HID + c], val);
  if (c == 0) {
    atomicAdd(&cnt_all[g], 1.f);
    if (view == 0) atomicAdd(&cntx[g], 1.f);
    else if (view == 1) atomicAdd(&cnty[g], 1.f);
  }
}

// ---------- build group embedding ge[g] = [sum_feat/valid , g_max] ----------
__global__ __launch_bounds__(256)
void build_ge_kernel(const unsigned* __restrict__ gmaxkey, const float* __restrict__ sumx,
                     const float* __restrict__ sumy, const float* __restrict__ cnt_all,
                     const float* __restrict__ cntx, const float* __restrict__ cnty,
                     float* __restrict__ ge) {
  const int idx = blockIdx.x * blockDim.x + threadIdx.x;
  if (idx >= N_GROUPS * HID) return;
  const int g = idx / HID;
  const int c = idx % HID;
  const float gm = (cnt_all[g] > 0.f) ? fdecode(gmaxkey[idx]) : 0.f;
  const float cx = cntx[g], cy = cnty[g];
  const float px = sumx[idx] / fmaxf(cx, 1.f);
  const float py = sumy[idx] / fmaxf(cy, 1.f);
  const float hx = (cx > 0.f) ? 1.f : 0.f;
  const float hy = (cy > 0.f) ? 1.f : 0.f;
  const float sf = px * hx + py * hy;
  const float valid = fmaxf(hx + hy, 1.f);
  ge[(size_t)g * (2 * HID) + c] = sf / valid;
  ge[(size_t)g * (2 * HID) + HID + c] = gm;
}

// ---------- pairwise scores: relu(A[i]+B[j]) @ W2 + b2 ----------
__global__ __launch_bounds__(256)
void scores_kernel(const float* __restrict__ A, const float* __restrict__ B,
                   const float* __restrict__ W2, const float* __restrict__ b2,
                   float* __restrict__ scores) {
  __shared__ float sW2[HID];
  if (threadIdx.x < HID) sW2[threadIdx.x] = W2[threadIdx.x];
  __syncthreads();
  const int idx = blockIdx.x * blockDim.x + threadIdx.x;
  if (idx >= N_GROUPS * N_GROUPS) return;
  const int i = idx / N_GROUPS;
  const int j = idx % N_GROUPS;
  const float* ai = A + (size_t)i * HID;
  const float* bj = B + (size_t)j * HID;
  float acc = b2[0];
#pragma unroll 4
  for (int c = 0; c < HID; ++c) acc += fmaxf(ai[c] + bj[c], 0.f) * sW2[c];
  scores[idx] = acc;
}

// ---------- symmetrize + sigmoid + event mask ----------
__global__ __launch_bounds__(256)
void final_kernel(const float* __restrict__ scores, const int* __restrict__ batch,
                  float* __restrict__ out) {
  const int idx = blockIdx.x * blockDim.x + threadIdx.x;
  if (idx >= N_GROUPS * N_GROUPS) return;
  const int i = idx / N_GROUPS;
  const int j = idx % N_GROUPS;
  const float sc = 0.5f * (scores[idx] + scores[(size_t)j * N_GROUPS + i]);
  out[idx] = (batch[i] == batch[j]) ? (1.f / (1.f + expf(-sc))) : 0.f;
}

// ---------- host orchestration ----------
extern "C" void kernel_launch(void* const* d_in, const int* in_sizes, int n_in,
                              void* d_out, int out_size, void* d_ws, size_t ws_size,
                              hipStream_t stream) {
  (void)in_sizes; (void)n_in; (void)out_size; (void)ws_size;
  const float* x       = (const float*)d_in[0];
  const int*   eidx    = (const int*)d_in[1];
  const float* eattr   = (const float*)d_in[2];
  const int*   gidx    = (const int*)d_in[3];
  const int*   batch   = (const int*)d_in[4];
  const float* embed_W = (const float*)d_in[5];
  const float* embed_b = (const float*)d_in[6];
  const float* Wqkv    = (const float*)d_in[7];
  const float* bqkv    = (const float*)d_in[8];
  const float* We      = (const float*)d_in[9];
  const float* be      = (const float*)d_in[10];
  const float* Wo      = (const float*)d_in[11];
  const float* bo      = (const float*)d_in[12];
  const float* ln1g    = (const float*)d_in[13];
  const float* ln1b    = (const float*)d_in[14];
  const float* W1      = (const float*)d_in[15];
  const float* b1      = (const float*)d_in[16];
  const float* W2      = (const float*)d_in[17];
  const float* b2      = (const float*)d_in[18];
  const float* ln2g    = (const float*)d_in[19];
  const float* ln2b    = (const float*)d_in[20];
  const float* affW1   = (const float*)d_in[21];
  const float* affb1   = (const float*)d_in[22];
  const float* affW2   = (const float*)d_in[23];
  const float* affb2   = (const float*)d_in[24];
  float* out = (float*)d_out;

  float* ws = (float*)d_ws;
  size_t off = 0;
  float* h      = ws + off; off += (size_t)N_HITS * HID;
  float* qkv    = ws + off; off += (size_t)N_HITS * 3 * HID;
  float* logits = ws + off; off += (size_t)N_EDGES * HEADS;
  unsigned* mkey = (unsigned*)(ws + off); off += (size_t)N_HITS * HEADS;
  float* ssum   = ws + off; off += (size_t)N_HITS * HEADS;   // ssum & attn contiguous (one fill)
  float* attn   = ws + off; off += (size_t)N_HITS * HID;
  float* tmpF   = ws + off; off += (size_t)N_HITS * FFN_DIM;
  float* tmpH   = ws + off; off += (size_t)N_HITS * HID;
  unsigned* gmaxkey = (unsigned*)(ws + off); off += (size_t)N_GROUPS * HID;
  float* sums   = ws + off; off += (size_t)(2 * N_GROUPS * HID + 3 * N_GROUPS);
  float* sumx = sums;
  float* sumy = sums + N_GROUPS * HID;
  float* cnt_all = sums + 2 * N_GROUPS * HID;
  float* cntx = cnt_all + N_GROUPS;
  float* cnty = cnt_all + 2 * N_GROUPS;
  float* ge     = ws + off; off += (size_t)N_GROUPS * 2 * HID;
  float* Aout   = ws + off; off += (size_t)N_GROUPS * HID;
  float* Bout   = ws + off; off += (size_t)N_GROUPS * HID;
  float* scores = ws + off; off += (size_t)N_GROUPS * N_GROUPS;

  const dim3 blk32(32);
  const int EH_BLOCKS = (N_EDGES * HEADS) / 256;        // 4096
  const int NH_BLOCKS = (N_HITS * HID) / 256;           // 8192
  const int GG_BLOCKS = (N_GROUPS * N_GROUPS) / 256;    // 1024

  // h = x @ embed_W + embed_b     (M=16384, K=25, N=128)
  gemm_wmma<IN_CH, HID, true, false>
      <<<dim3(N_HITS / 32, HID / 64), blk32, 0, stream>>>(x, embed_W, embed_b, h);

  for (int l = 0; l < LAYERS; ++l) {
    const float* lWqkv = Wqkv + (size_t)l * HID * 3 * HID;
    const float* lbqkv = bqkv + (size_t)l * 3 * HID;
    const float* lWe   = We   + (size_t)l * EDGE_DIM * HID;
    const float* lbe   = be   + (size_t)l * HID;
    const float* lWo   = Wo   + (size_t)l * HID * HID;
    const float* lbo   = bo   + (size_t)l * HID;
    const float* lW1   = W1   + (size_t)l * HID * FFN_DIM;
    const float* lb1   = b1   + (size_t)l * FFN_DIM;
    const float* lW2   = W2   + (size_t)l * FFN_DIM * HID;
    const float* lb2   = b2   + (size_t)l * HID;

    // qkv = h @ Wqkv + bqkv       (K=128, N=384)
    gemm_wmma<HID, 3 * HID, true, false>
        <<<dim3(N_HITS / 32, (3 * HID) / 64), blk32, 0, stream>>>(h, lWqkv, lbqkv, qkv);

    {
      int n = N_HITS * HEADS;
      fill_u32<<<(n + 255) / 256, 256, 0, stream>>>(mkey, FKEY_NEG_INF, n);
      int n2 = N_HITS * (HEADS + HID);  // ssum + attn zero
      fill_u32<<<(n2 + 255) / 256, 256, 0, stream>>>((unsigned*)ssum, 0u, n2);
    }

    edge_logits_kernel<<<EH_BLOCKS, 256, 0, stream>>>(eidx, eattr, qkv, lWe, lbe, logits);
    edge_segmax_kernel<<<EH_BLOCKS, 256, 0, stream>>>(eidx, logits, mkey);
    edge_accum_kernel<<<EH_BLOCKS, 256, 0, stream>>>(eidx, eattr, qkv, lWe, lbe,
                                                     logits, mkey, ssum, attn);
    attn_div_kernel<<<NH_BLOCKS, 256, 0, stream>>>(attn, ssum);

    // tmpH = attn @ Wo + bo       (K=128, N=128)
    gemm_wmma<HID, HID, true, false>
        <<<dim3(N_HITS / 32, HID / 64), blk32, 0, stream>>>(attn, lWo, lbo, tmpH);
    add_ln_kernel<<<N_HITS, HID, 0, stream>>>(h, tmpH, ln1g + l * HID, ln1b + l * HID);

    // tmpF = relu(h @ W1 + b1)    (K=128, N=512)
    gemm_wmma<HID, FFN_DIM, true, true>
        <<<dim3(N_HITS / 32, FFN_DIM / 64), blk32, 0, stream>>>(h, lW1, lb1, tmpF);
    // tmpH = tmpF @ W2 + b2       (K=512, N=128)
    gemm_wmma<FFN_DIM, HID, true, false>
        <<<dim3(N_HITS / 32, HID / 64), blk32, 0, stream>>>(tmpF, lW2, lb2, tmpH);
    add_ln_kernel<<<N_HITS, HID, 0, stream>>>(h, tmpH, ln2g + l * HID, ln2b + l * HID);
  }

  // group pooling
  {
    int n = N_GROUPS * HID;
    fill_u32<<<(n + 255) / 256, 256, 0, stream>>>(gmaxkey, FKEY_NEG_INF, n);
    int n2 = 2 * N_GROUPS * HID + 3 * N_GROUPS;
    fill_u32<<<(n2 + 255) / 256, 256, 0, stream>>>((unsigned*)sums, 0u, n2);
  }
  pool_scatter_kernel<<<NH_BLOCKS, 256, 0, stream>>>(h, x, gidx, gmaxkey,
                                                     sumx, sumy, cnt_all, cntx, cnty);
  build_ge_kernel<<<(N_GROUPS * HID) / 256, 256, 0, stream>>>(gmaxkey, sumx, sumy,
                                                              cnt_all, cntx, cnty, ge);

  // affinity head: Aout = ge @ W1a + b1 ; Bout = ge @ W1b   (M=512, K=256, N=128)
  gemm_wmma<2 * HID, HID, true, false>
      <<<dim3(N_GROUPS / 32, HID / 64), blk32, 0, stream>>>(ge, affW1, affb1, Aout);
  gemm_wmma<2 * HID, HID, false, false>
      <<<dim3(N_GROUPS / 32, HID / 64), blk32, 0, stream>>>(ge, affW1 + (size_t)2 * HID * HID,
                                                            nullptr, Bout);
  scores_kernel<<<GG_BLOCKS, 256, 0, stream>>>(Aout, Bout, affW2, affb2, scores);
  final_kernel<<<GG_BLOCKS, 256, 0, stream>>>(scores, batch, out);
}